// LSTM_43155831390875
// MI455X (gfx1250) — compile-verified
//
#include <hip/hip_runtime.h>

// Problem constants from the reference: B=256, S=512, V=50000, E=100, H=512.
constexpr int Bsz  = 256;
constexpr int Slen = 512;
constexpr int Edim = 100;
constexpr int Hdim = 512;
constexpr int KTOT = Edim + Hdim; // 612; E=100 is 4-aligned so chunks never straddle

typedef __attribute__((ext_vector_type(2))) float v2f;
typedef __attribute__((ext_vector_type(8))) float v8f;

__device__ __forceinline__ float sigmoidf(float x) {
    return 1.0f / (1.0f + __expf(-x));
}

__global__ __launch_bounds__(256) void lstm_zero_init(float* __restrict__ h0,
                                                      float* __restrict__ c0,
                                                      int n) {
    int i = blockIdx.x * blockDim.x + threadIdx.x;
    if (i < n) { h0[i] = 0.0f; c0[i] = 0.0f; }
}

// One LSTM timestep.
// Grid: (B/32, H/64). Block: 128 threads = 4 waves.
// Each wave computes TWO 16x16 M-tiles (batch rows m_base..m_base+31) for each
// of the 4 gate quadrants (i,f,g,o) via V_WMMA_F32_16X16X4_F32, sharing every
// B (weight) load across both M-tiles, then applies the cell update locally.
//
// WMMA f32 16x16x4 VGPR striping (ISA 7.12.2):
//   A (16x4): lanes 0-15 -> M=lane, VGPR0=K(2*half+0), VGPR1=K(2*half+1)
//   B (4x16): lane -> N=lane&15, same K pairing -> contiguous K pair per lane
//   C (16x16): VGPR r, lane l -> M = r + 8*(l>>4), N = l&15
__global__ __launch_bounds__(128) void lstm_step(
    const int*   __restrict__ sentence,  // [B, S] (token ids)
    const float* __restrict__ emb,       // [V, E]
    const float* __restrict__ Wih,       // [4H, E] row-major (K contiguous)
    const float* __restrict__ Whh,       // [4H, H] row-major (K contiguous)
    const float* __restrict__ bih,       // [4H]
    const float* __restrict__ bhh,       // [4H]
    const float* __restrict__ h_in,      // [B, H]
    float*       __restrict__ c,         // [B, H], updated in place
    float*       __restrict__ h_out,     // [B, H]
    int t)
{
    // Two k-major panels (stride 16 floats -> conflict-free WMMA A fetches):
    // panel p holds batch rows m_base + p*16 .. m_base + p*16 + 15.
    __shared__ float Alds[2][KTOT * 16]; // 2 * 39168 B = 78336 B

    const int tid    = threadIdx.x;
    const int m_base = blockIdx.x * 32;

    // ---- Cooperative A-panel build: A = [ emb[sentence[b,t]] | h_in[b] ] ----
    {
        const int r   = tid >> 2;            // row 0..31
        const int sub = tid & 3;             // 4 threads per row
        const int b   = m_base + r;
        float* panel  = Alds[r >> 4];
        const int rr  = r & 15;
        const int tok = sentence[(long)b * Slen + t];
        const float* erow = emb + (long)tok * Edim;
        for (int k = sub; k < Edim; k += 4)
            panel[k * 16 + rr] = erow[k];
        const float* hrow = h_in + (long)b * Hdim;
        for (int k = sub; k < Hdim; k += 4)
            panel[(Edim + k) * 16 + rr] = hrow[k];
    }
    __syncthreads();

    const int wave = tid >> 5;               // 0..3
    const int lane = tid & 31;
    const int half = lane >> 4;              // selects K pair (0/1 vs 2/3)
    const int ml   = lane & 15;              // M for A, N for B/C
    const int col  = blockIdx.y * 64 + wave * 16 + ml;  // hidden column 0..511

    // acc[mt][q]: M-tile mt (0/1), gate quadrant q (i,f,g,o)
    v8f acc[2][4];
#pragma unroll
    for (int mt = 0; mt < 2; ++mt)
#pragma unroll
        for (int q = 0; q < 4; ++q)
            acc[mt][q] = (v8f){};

    // Weight row bases for the 4 gate quadrants (PyTorch order i,f,g,o).
    const float* wi0 = Wih + (long)(0 * Hdim + col) * Edim;
    const float* wi1 = Wih + (long)(1 * Hdim + col) * Edim;
    const float* wi2 = Wih + (long)(2 * Hdim + col) * Edim;
    const float* wi3 = Wih + (long)(3 * Hdim + col) * Edim;

    // ---- K loop, x part (K = 0..E) ----
    for (int k0 = 0; k0 < Edim; k0 += 4) {
        const int kb = k0 + 2 * half;        // even -> 8B-aligned B loads
        v2f a0, a1;
        a0.x = Alds[0][kb * 16 + ml];
        a0.y = Alds[0][(kb + 1) * 16 + ml];
        a1.x = Alds[1][kb * 16 + ml];
        a1.y = Alds[1][(kb + 1) * 16 + ml];
        v2f b0 = *(const v2f*)(wi0 + kb);
        v2f b1 = *(const v2f*)(wi1 + kb);
        v2f b2 = *(const v2f*)(wi2 + kb);
        v2f b3 = *(const v2f*)(wi3 + kb);
        acc[0][0] = __builtin_amdgcn_wmma_f32_16x16x4_f32(false, a0, false, b0, (short)0, acc[0][0], false, false);
        acc[1][0] = __builtin_amdgcn_wmma_f32_16x16x4_f32(false, a1, false, b0, (short)0, acc[1][0], false, false);
        acc[0][1] = __builtin_amdgcn_wmma_f32_16x16x4_f32(false, a0, false, b1, (short)0, acc[0][1], false, false);
        acc[1][1] = __builtin_amdgcn_wmma_f32_16x16x4_f32(false, a1, false, b1, (short)0, acc[1][1], false, false);
        acc[0][2] = __builtin_amdgcn_wmma_f32_16x16x4_f32(false, a0, false, b2, (short)0, acc[0][2], false, false);
        acc[1][2] = __builtin_amdgcn_wmma_f32_16x16x4_f32(false, a1, false, b2, (short)0, acc[1][2], false, false);
        acc[0][3] = __builtin_amdgcn_wmma_f32_16x16x4_f32(false, a0, false, b3, (short)0, acc[0][3], false, false);
        acc[1][3] = __builtin_amdgcn_wmma_f32_16x16x4_f32(false, a1, false, b3, (short)0, acc[1][3], false, false);
    }

    const float* wh0 = Whh + (long)(0 * Hdim + col) * Hdim;
    const float* wh1 = Whh + (long)(1 * Hdim + col) * Hdim;
    const float* wh2 = Whh + (long)(2 * Hdim + col) * Hdim;
    const float* wh3 = Whh + (long)(3 * Hdim + col) * Hdim;

    // ---- K loop, h part (K = E..E+H) ----
    for (int k0 = 0; k0 < Hdim; k0 += 4) {
        const int kb = k0 + 2 * half;
        v2f a0, a1;
        a0.x = Alds[0][(Edim + kb) * 16 + ml];
        a0.y = Alds[0][(Edim + kb + 1) * 16 + ml];
        a1.x = Alds[1][(Edim + kb) * 16 + ml];
        a1.y = Alds[1][(Edim + kb + 1) * 16 + ml];
        v2f b0 = *(const v2f*)(wh0 + kb);
        v2f b1 = *(const v2f*)(wh1 + kb);
        v2f b2 = *(const v2f*)(wh2 + kb);
        v2f b3 = *(const v2f*)(wh3 + kb);
        acc[0][0] = __builtin_amdgcn_wmma_f32_16x16x4_f32(false, a0, false, b0, (short)0, acc[0][0], false, false);
        acc[1][0] = __builtin_amdgcn_wmma_f32_16x16x4_f32(false, a1, false, b0, (short)0, acc[1][0], false, false);
        acc[0][1] = __builtin_amdgcn_wmma_f32_16x16x4_f32(false, a0, false, b1, (short)0, acc[0][1], false, false);
        acc[1][1] = __builtin_amdgcn_wmma_f32_16x16x4_f32(false, a1, false, b1, (short)0, acc[1][1], false, false);
        acc[0][2] = __builtin_amdgcn_wmma_f32_16x16x4_f32(false, a0, false, b2, (short)0, acc[0][2], false, false);
        acc[1][2] = __builtin_amdgcn_wmma_f32_16x16x4_f32(false, a1, false, b2, (short)0, acc[1][2], false, false);
        acc[0][3] = __builtin_amdgcn_wmma_f32_16x16x4_f32(false, a0, false, b3, (short)0, acc[0][3], false, false);
        acc[1][3] = __builtin_amdgcn_wmma_f32_16x16x4_f32(false, a1, false, b3, (short)0, acc[1][3], false, false);
    }

    // ---- Cell update (each (row, col) owned by exactly one lane) ----
    const float bs0 = bih[0 * Hdim + col] + bhh[0 * Hdim + col];
    const float bs1 = bih[1 * Hdim + col] + bhh[1 * Hdim + col];
    const float bs2 = bih[2 * Hdim + col] + bhh[2 * Hdim + col];
    const float bs3 = bih[3 * Hdim + col] + bhh[3 * Hdim + col];

#pragma unroll
    for (int mt = 0; mt < 2; ++mt) {
#pragma unroll
        for (int r = 0; r < 8; ++r) {
            const int  mg  = m_base + mt * 16 + r + 8 * half;
            const long idx = (long)mg * Hdim + col;
            const float ig = sigmoidf(acc[mt][0][r] + bs0);
            const float fg = sigmoidf(acc[mt][1][r] + bs1);
            const float gg = tanhf   (acc[mt][2][r] + bs2);
            const float og = sigmoidf(acc[mt][3][r] + bs3);
            const float cn = fg * c[idx] + ig * gg;
            c[idx]     = cn;
            h_out[idx] = og * tanhf(cn);
        }
    }
}

extern "C" void kernel_launch(void* const* d_in, const int* in_sizes, int n_in,
                              void* d_out, int out_size, void* d_ws, size_t ws_size,
                              hipStream_t stream) {
    const int*   sentence = (const int*)  d_in[0];  // [B,S] token ids
    const float* emb      = (const float*)d_in[1];  // [V,E]
    const float* Wih      = (const float*)d_in[2];  // [4H,E]
    const float* Whh      = (const float*)d_in[3];  // [4H,H]
    const float* bih      = (const float*)d_in[4];  // [4H]
    const float* bhh      = (const float*)d_in[5];  // [4H]
    float* out = (float*)d_out;                     // [B,H]
    float* ws  = (float*)d_ws;

    const int BH = Bsz * Hdim;                      // 131072 floats = 512 KB
    float* hbuf0 = ws;                              // h ping
    float* hbuf1 = ws + BH;                         // h pong
    float* cbuf  = ws + 2 * BH;                     // c state (in-place)

    lstm_zero_init<<<(BH + 255) / 256, 256, 0, stream>>>(hbuf0, cbuf, BH);

    dim3 grid(Bsz / 32, Hdim / 64);                 // (8, 8)
    for (int t = 0; t < Slen; ++t) {
        const float* hin  = (t & 1) ? hbuf1 : hbuf0;
        float*       hout = (t == Slen - 1) ? out
                          : ((t & 1) ? hbuf0 : hbuf1);
        lstm_step<<<grid, 128, 0, stream>>>(sentence, emb, Wih, Whh, bih, bhh,
                                            hin, cbuf, hout, t);
    }
}